// EntropySpatLoss_18769007083963
// MI455X (gfx1250) — compile-verified
//
#include <hip/hip_runtime.h>

// ---------------------------------------------------------------------------
// EntropySpatLoss for MI455X (gfx1250, wave32).
//
// Sizes (fixed by the reference): B=8, N=65536, P=200, C=20.
// acts: [B,N,P] f32 (419 MB)  labels: [B,N] int  ident: [P,C] f32.
//
// Design: single HBM pass (419 MB ~= 18 us at 23.3 TB/s) with online softmax
// (m,Z,S per (segment, prototype)); blocks specialized per (segment, chunk)
// so the class index is uniform and accumulators stay in VGPRs. Row fetch is
// vectorized: b128 + b64 + b32 per pixel (3 VMEM instructions for 800 B).
// Final tiny einsum done with V_WMMA_F32_16X16X4_F32.
// ---------------------------------------------------------------------------

#define B_    8
#define N_    65536
#define P_    200
#define C_    20
#define NSEG  (B_ * C_)     // 160
#define NCH   8             // chunks per segment (fan-in of the merge)
#define WAVES 8
#define BLK   (WAVES * 32)  // 256 threads

typedef float v2f __attribute__((ext_vector_type(2)));
typedef float v4f __attribute__((ext_vector_type(4)));
typedef float v8f __attribute__((ext_vector_type(8)));

// Lane -> prototype slot mapping (7 slots per lane, contiguous vector loads):
//   j = 0..3 : p = 4*lane + j            (protos   0..127, one b128)
//   j = 4..5 : p = 128 + 2*lane + (j-4)  (protos 128..191, one b64)
//   j = 6    : p = 192 + lane, lane < 8  (protos 192..199, one b32)
__device__ __forceinline__ int slot_proto(int lane, int j) {
  if (j < 4)  return 4 * lane + j;
  if (j < 6)  return 128 + 2 * lane + (j - 4);
  return 192 + lane;   // valid only for lane < 8
}

// ---------------------------------------------------------------------------
// Kernel 1: streaming online-softmax accumulation.
// grid = (NSEG, NCH); block = 256. Block (s, ch) scans pixels
// [ch*N/NCH, (ch+1)*N/NCH) of image b = s/C, keeps only label-1==c = s%C.
// ---------------------------------------------------------------------------
__global__ __launch_bounds__(BLK)
void seg_online_kernel(const float* __restrict__ acts,
                       const int*   __restrict__ labels,
                       int*         __restrict__ hist,
                       float*       __restrict__ partials) {
  const int s    = blockIdx.x;          // segment 0..159
  const int ch   = blockIdx.y;          // chunk   0..NCH-1
  const int b    = s / C_;
  const int c    = s % C_;
  const int lane = threadIdx.x & 31;
  const int w    = threadIdx.x >> 5;

  float m[7], Z[7], S[7];
#pragma unroll
  for (int j = 0; j < 7; ++j) { m[j] = -1e30f; Z[j] = 0.f; S[j] = 0.f; }

  const int chunk0 = ch * (N_ / NCH);
  const int chunk1 = chunk0 + (N_ / NCH);
  const int*   lb   = labels + b * N_;
  const float* base = acts + (size_t)b * N_ * P_;

  int matched = 0;  // wave-uniform
  for (int g = chunk0 + w * 32; g < chunk1; g += WAVES * 32) {
    const int lab  = lb[g + lane];
    const bool hit = (lab - 1 == c);
    unsigned mask  = __builtin_amdgcn_ballot_w32(hit);
    matched += __popc(mask);
    while (mask) {
      const int src = __ffs(mask) - 1;
      mask &= mask - 1u;
      if (mask) {  // prefetch the next matched row (global_prefetch_b8)
        const int nx = __ffs(mask) - 1;
        __builtin_prefetch(base + (size_t)(g + nx) * P_ + lane * 8, 0, 1);
      }
      const float* row = base + (size_t)(g + src) * P_;  // 800 B, 16 B aligned

      // 3 coalesced vector loads per pixel (512 B + 256 B + 32 B)
      const v4f   x4 = *(const v4f*)(row + 4 * lane);
      const v2f   x2 = *(const v2f*)(row + 128 + 2 * lane);
      const float x1 = row[192 + (lane & 7)];   // lanes >= 8 load a dummy copy
      float xs[7] = {x4.x, x4.y, x4.z, x4.w, x2.x, x2.y, x1};

#pragma unroll
      for (int j = 0; j < 7; ++j) {
        const float x  = xs[j];
        const float mn = fmaxf(m[j], x);
        const float dm = m[j] - mn;                   // <= 0
        const float dx = x - mn;                      // <= 0
        const float f  = __expf(dm);
        const float d  = __expf(dx);
        const float t  = fmaf(dm, Z[j], S[j]);        // S + dm*Z (old Z)
        Z[j] = fmaf(Z[j], f, d);
        S[j] = fmaf(dx, d, t * f);
        m[j] = mn;
      }
    }
  }

  // wave partials -> LDS, then block-level merge over the 8 waves.
  __shared__ float lsm[WAVES][3][P_];   // 19.2 KB
#pragma unroll
  for (int j = 0; j < 7; ++j) {
    const int p = slot_proto(lane, j);
    if (j < 6 || lane < 8) { lsm[w][0][p] = m[j]; lsm[w][1][p] = Z[j]; lsm[w][2][p] = S[j]; }
  }
  if (lane == 0 && matched) atomicAdd(&hist[s], matched);  // exact int atomics
  __syncthreads();

  for (int p = threadIdx.x; p < P_; p += BLK) {   // threads 0..199
    float M = -1e30f;
#pragma unroll
    for (int wv = 0; wv < WAVES; ++wv) M = fmaxf(M, lsm[wv][0][p]);
    float Zt = 0.f, St = 0.f;
#pragma unroll
    for (int wv = 0; wv < WAVES; ++wv) {
      const float mw = lsm[wv][0][p];
      const float f  = __expf(mw - M);
      Zt = fmaf(lsm[wv][1][p], f, Zt);
      St = fmaf(f, fmaf(mw - M, lsm[wv][1][p], lsm[wv][2][p]), St);
    }
    float* outp = partials + (size_t)(s * NCH + ch) * (3 * P_);
    outp[p] = M; outp[P_ + p] = Zt; outp[2 * P_ + p] = St;
  }
}

// ---------------------------------------------------------------------------
// Kernel 2: merge the NCH chunk-partials per (segment, prototype), compute the
// normalized entropy ent[s][p] = (logZ - S/Z) / log(count), 0 if count < 2.
// ---------------------------------------------------------------------------
__global__ void merge_entropy_kernel(const float* __restrict__ partials,
                                     const int*   __restrict__ hist,
                                     float*       __restrict__ ent) {
  const int idx = blockIdx.x * blockDim.x + threadIdx.x;
  if (idx >= NSEG * P_) return;
  const int s = idx / P_;
  const int p = idx % P_;

  float M = -1e30f, Z = 0.f, S = 0.f;
  const float* basep = partials + (size_t)s * NCH * 3 * P_;
  for (int ch = 0; ch < NCH; ++ch) {
    const float* pp = basep + ch * 3 * P_;
    const float m2 = pp[p], z2 = pp[P_ + p], s2 = pp[2 * P_ + p];
    const float Mn = fmaxf(M, m2);
    const float f1 = __expf(M - Mn);
    const float f2 = __expf(m2 - Mn);
    const float Zn = Z * f1 + z2 * f2;
    const float Sn = f1 * fmaf(M - Mn, Z, S) + f2 * fmaf(m2 - Mn, z2, s2);
    M = Mn; Z = Zn; S = Sn;
  }
  const int cnt = hist[s];
  float val = 0.f;
  if (cnt >= 2) {
    const float Zc = fmaxf(Z, 1e-30f);
    val = (__logf(Zc) - S / Zc) / __logf((float)cnt);
  }
  ent[idx] = val;
}

// ---------------------------------------------------------------------------
// Kernel 3: per-image ent_cls[b,c] = sum_p ENT[b,c,p]*IDENT[p,c] via
// V_WMMA_F32_16X16X4_F32 (diagonal of the 20x20 product, two 16x16 tiles,
// K-loop over 200), then masked mean -> scalar loss.
// Single wave32 block: EXEC is all ones at every WMMA.
// ---------------------------------------------------------------------------
__global__ __launch_bounds__(32)
void finalize_wmma_kernel(const float* __restrict__ ent,
                          const int*   __restrict__ hist,
                          const float* __restrict__ ident,
                          float*       __restrict__ out) {
  const int lane = threadIdx.x;      // 0..31
  const int l15  = lane & 15;
  const int koff = (lane < 16) ? 0 : 2;   // A/B lane-half K offset (ISA 16x16x4 layout)

  __shared__ float tile[16][16];
  __shared__ float entcls[C_];
  __shared__ float pcnt[C_];

  if (lane < C_) {
    float sum = 0.f;
    for (int p = 0; p < P_; ++p) sum += ident[p * C_ + lane];
    pcnt[lane] = sum;
  }
  __syncthreads();

  float lossSum = 0.f;
  int   nvalid  = 0;
  for (int b = 0; b < B_; ++b) {
    for (int t = 0; t < 2; ++t) {            // diag tiles (0..15,0..15) and (16..19,16..19)
      const int roff = t * 16, coff = t * 16;
      v8f acc = {0.f, 0.f, 0.f, 0.f, 0.f, 0.f, 0.f, 0.f};

      const int   row    = roff + l15;
      const float row_ok = (row < C_) ? 1.0f : 0.0f;
      const int   rowc   = (row < C_) ? row : 0;
      const int   col    = coff + l15;
      const float col_ok = (col < C_) ? 1.0f : 0.0f;
      const int   colc   = (col < C_) ? col : 0;
      const float* arow  = ent + (size_t)(b * C_ + rowc) * P_;

      for (int k0 = 0; k0 < P_; k0 += 4) {
        // A[M=row][K=k0+koff+{0,1}]  (16x4 f32 tile, 2 VGPRs/lane)
        v2f a;
        a.x = arow[k0 + koff + 0] * row_ok;
        a.y = arow[k0 + koff + 1] * row_ok;
        // B[K=k0+koff+{0,1}][N=col] (4x16 f32 tile, 2 VGPRs/lane)
        v2f bb;
        bb.x = ident[(k0 + koff + 0) * C_ + colc] * col_ok;
        bb.y = ident[(k0 + koff + 1) * C_ + colc] * col_ok;
        acc = __builtin_amdgcn_wmma_f32_16x16x4_f32(
            false, a, false, bb, (short)0, acc, false, false);
      }
      // D layout: VGPR r = D[r + (lane<16 ? 0 : 8)][l15]
#pragma unroll
      for (int r = 0; r < 8; ++r) {
        const int Mrow = r + ((lane < 16) ? 0 : 8);
        tile[Mrow][l15] = acc[r];
      }
      __syncthreads();
      if (lane < 16) {
        const int cg = coff + lane;
        if (cg < C_) entcls[cg] = tile[lane][lane];
      }
      __syncthreads();
    }
    if (lane == 0) {
      for (int c = 0; c < C_; ++c) {
        const bool valid = (hist[b * C_ + c] >= 2) && (pcnt[c] > 0.f);
        if (valid) {
          lossSum += entcls[c] / fmaxf(pcnt[c], 1.0f);
          nvalid  += 1;
        }
      }
    }
    __syncthreads();
  }
  if (lane == 0) out[0] = (nvalid > 0) ? (lossSum / (float)nvalid) : 0.f;
}

// ---------------------------------------------------------------------------
// Host launcher. ws layout: [hist 160 ints][pad to 1KB][partials 160*8*600 f32]
// [ent 32000 f32]  (~3.2 MB total).
// ---------------------------------------------------------------------------
extern "C" void kernel_launch(void* const* d_in, const int* in_sizes, int n_in,
                              void* d_out, int out_size, void* d_ws, size_t ws_size,
                              hipStream_t stream) {
  (void)in_sizes; (void)n_in; (void)out_size; (void)ws_size;

  const float* acts   = (const float*)d_in[0];   // [B,N,P] f32
  const int*   labels = (const int*)d_in[1];     // [B,N]   int
  const float* ident  = (const float*)d_in[2];   // [P,C]   f32
  float*       outp   = (float*)d_out;

  char*  ws       = (char*)d_ws;
  int*   hist     = (int*)ws;
  float* partials = (float*)(ws + 1024);
  float* ent      = (float*)(ws + 1024 + (size_t)NSEG * NCH * 3 * P_ * sizeof(float));

  hipMemsetAsync(hist, 0, NSEG * sizeof(int), stream);

  dim3 g1(NSEG, NCH);
  seg_online_kernel<<<g1, BLK, 0, stream>>>(acts, labels, hist, partials);

  merge_entropy_kernel<<<(NSEG * P_ + 255) / 256, 256, 0, stream>>>(partials, hist, ent);

  finalize_wmma_kernel<<<1, 32, 0, stream>>>(ent, hist, ident, outp);
}